// PAGTNLayer_80333068304729
// MI455X (gfx1250) — compile-verified
//
#include <hip/hip_runtime.h>
#include <hip/hip_bf16.h>
#include <math.h>

// ---------------------------------------------------------------------------
// PAGTN layer for MI455X (gfx1250, wave32).
// Dense 64x64 projections via V_WMMA_F32_16X16X4_F32 (fp32-exact matrix path),
// per-edge attention with int-ordered atomics for segment max and f32 atomicAdd for sum.
// ---------------------------------------------------------------------------

typedef float v2f __attribute__((ext_vector_type(2)));
typedef float v8f __attribute__((ext_vector_type(8)));

__device__ __forceinline__ v8f wmma_f32_4(v2f a, v2f b, v8f c) {
  // 8 args: (neg_a, A, neg_b, B, c_mod, C, reuse_a, reuse_b)
  return __builtin_amdgcn_wmma_f32_16x16x4_f32(false, a, false, b, (short)0, c,
                                               false, false);
}

__device__ __forceinline__ void atomicMaxFloat(float* addr, float val) {
  // Works for all-sign floats given init to -inf (0xFF800000).
  if (val >= 0.0f) atomicMax((int*)addr, __float_as_int(val));
  else             atomicMin((unsigned int*)addr, __float_as_uint(val));
}

#define XPAD 68   // LDS X-tile row stride (floats): 16B-aligned b128, bank-spread
#define WPAD 80   // LDS paired-W row stride (float2): lane-halves hit disjoint banks

// ---------------------------------------------------------------------------
// Fused projection: O_j = X @ W_j + b_j for up to 4 weight matrices.
// X: (R, 64) row-major, W: (64, 64) row-major [fan_in x fan_out], b: (64,).
// One wave computes a 16-row x 64-col strip; block = 8 waves = 128 rows.
// Weights staged into LDS as K-pairs so each B fragment is one ds_load_b64.
// ---------------------------------------------------------------------------
__global__ __launch_bounds__(256) void pagtn_proj64_kernel(
    const float* __restrict__ X, int R,
    const float* __restrict__ W0, const float* __restrict__ B0, float* __restrict__ O0,
    const float* __restrict__ W1, const float* __restrict__ B1, float* __restrict__ O1,
    const float* __restrict__ W2, const float* __restrict__ B2, float* __restrict__ O2,
    const float* __restrict__ W3, const float* __restrict__ B3, float* __restrict__ O3,
    int nmat)
{
  __shared__ __align__(16) float sWp[32 * WPAD * 2]; // 20 KB: (W[2kk][n], W[2kk+1][n])
  __shared__ float sB[64];
  __shared__ __align__(16) float sX[8][16 * XPAD];   // ~34 KB (per-wave X tile)

  const float* Ws[4] = {W0, W1, W2, W3};
  const float* Bs[4] = {B0, B1, B2, B3};
  float*       Os[4] = {O0, O1, O2, O3};

  const int t    = threadIdx.x;
  const int wave = t >> 5;
  const int lane = t & 31;
  const int m    = lane & 15;   // row within 16-row strip (A) / col within N-tile (B,D)
  const int h    = lane >> 4;   // lane half: selects K-pair (A,B) / M-half (D)
  const int row0 = blockIdx.x * 128 + wave * 16;

  // Stage this wave's 16x64 X tile into LDS (zero-pad OOB rows), b128 stores.
  for (int i = lane; i < 16 * 16; i += 32) {
    const int r   = i >> 4;
    const int c4  = i & 15;
    const int row = row0 + r;
    float4 v;
    if (row < R) v = ((const float4*)X)[row * 16 + c4];
    else { v.x = 0.f; v.y = 0.f; v.z = 0.f; v.w = 0.f; }
    *(float4*)&sX[wave][r * XPAD + c4 * 4] = v;
  }

#pragma unroll
  for (int j = 0; j < 4; ++j) {
    if (j >= nmat) break;
    __syncthreads();  // previous sWp consumers done (also covers X staging on j==0)
    // Pair K-adjacent weight rows: sWp[kk][n] = {W[2kk][n], W[2kk+1][n]}.
    const float* Wj = Ws[j];
    for (int i = t; i < 32 * 64; i += 256) {
      const int kk = i >> 6;   // K-pair index 0..31
      const int n  = i & 63;   // output column
      v2f w2;
      w2.x = Wj[(2 * kk + 0) * 64 + n];
      w2.y = Wj[(2 * kk + 1) * 64 + n];
      *(v2f*)&sWp[(kk * WPAD + n) * 2] = w2;
    }
    if (t < 64) sB[t] = Bs[j][t];
    __syncthreads();

    v8f acc[4] = {};  // N-tiles 0..3 of the 16x64 output strip
    const float* sXw = sX[wave];
#pragma unroll
    for (int k0 = 0; k0 < 64; k0 += 4) {
      const int k  = k0 + h * 2;            // A/B: K = 2*half + vgpr
      const int kk = (k0 >> 1) + h;         // K-pair index for this lane half
      v2f a = *(const v2f*)&sXw[m * XPAD + k];
#pragma unroll
      for (int nt = 0; nt < 4; ++nt) {
        const int n = nt * 16 + m;
        v2f b = *(const v2f*)&sWp[(kk * WPAD + n) * 2];
        acc[nt] = wmma_f32_4(a, b, acc[nt]);
      }
    }

    // D layout: VGPR jv holds M = jv + 8*h, N = lane%16 (+16*nt). Add bias, store.
    float* Oj = Os[j];
#pragma unroll
    for (int jv = 0; jv < 8; ++jv) {
      const int row = row0 + jv + 8 * h;
      if (row < R) {
#pragma unroll
        for (int nt = 0; nt < 4; ++nt) {
          const int n = nt * 16 + m;
          Oj[row * 64 + n] = acc[nt][jv] + sB[n];
        }
      }
    }
  }
}

// ---------------------------------------------------------------------------
// Init segment-max buffer to -inf and segment-sum buffer to 0.
// ---------------------------------------------------------------------------
__global__ void pagtn_init_kernel(float* __restrict__ mbuf,
                                  float* __restrict__ denom, int n) {
  const int i = blockIdx.x * blockDim.x + threadIdx.x;
  if (i < n) { mbuf[i] = -__builtin_inff(); denom[i] = 0.0f; }
}

// ---------------------------------------------------------------------------
// scores[e,h] = relu(s[src,h,:] + d[dst,h,:] + eatn[e,:]) . Wdot + bdot
// Also: segment max into mbuf[dst,h] via atomic ordering trick.
// ---------------------------------------------------------------------------
__global__ __launch_bounds__(256) void pagtn_scores_kernel(
    const float* __restrict__ s, const float* __restrict__ d,
    const float* __restrict__ eatn,
    const int* __restrict__ src, const int* __restrict__ dst,
    const float* __restrict__ Wdot, const float* __restrict__ bdot,
    float* __restrict__ scores, float* __restrict__ mbuf, int EH)
{
  __shared__ __align__(16) float sw[64];
  __shared__ float sb0;
  if (threadIdx.x < 64) sw[threadIdx.x] = Wdot[threadIdx.x];
  if (threadIdx.x == 0) sb0 = bdot[0];
  __syncthreads();

  const int idx = blockIdx.x * blockDim.x + threadIdx.x;
  if (idx >= EH) return;
  const int e  = idx >> 2;
  const int hh = idx & 3;
  const int se = src[e], de = dst[e];

  const float4* sp = (const float4*)(s + (size_t)(se * 4 + hh) * 64);
  const float4* dp = (const float4*)(d + (size_t)(de * 4 + hh) * 64);
  const float4* ep = (const float4*)(eatn + (size_t)e * 64);
  const float4* wp = (const float4*)sw;

  float acc = sb0;
#pragma unroll 4
  for (int k = 0; k < 16; ++k) {
    const float4 a = sp[k], b = dp[k], c = ep[k], w = wp[k];
    acc += fmaxf(a.x + b.x + c.x, 0.f) * w.x
         + fmaxf(a.y + b.y + c.y, 0.f) * w.y
         + fmaxf(a.z + b.z + c.z, 0.f) * w.z
         + fmaxf(a.w + b.w + c.w, 0.f) * w.w;
  }
  scores[idx] = acc;
  atomicMaxFloat(&mbuf[de * 4 + hh], acc);
}

// ---------------------------------------------------------------------------
// ex = exp(score - max[dst,h]); scores <- ex; denom[dst,h] += ex (atomic).
// ---------------------------------------------------------------------------
__global__ __launch_bounds__(256) void pagtn_exsum_kernel(
    const int* __restrict__ dst, const float* __restrict__ mbuf,
    float* __restrict__ scores, float* __restrict__ denom, int EH)
{
  const int idx = blockIdx.x * blockDim.x + threadIdx.x;
  if (idx >= EH) return;
  const int e  = idx >> 2;
  const int hh = idx & 3;
  const int de = dst[e];
  const float ex = __expf(scores[idx] - mbuf[de * 4 + hh]);
  scores[idx] = ex;
  atomicAdd(&denom[de * 4 + hh], ex);
}

// ---------------------------------------------------------------------------
// out[dst,h,:] += (ex/denom[dst,h]) * (md[src,h,:] + medg[e,:])
// (out already holds node_feats @ Wn + bn from the projection kernel)
// ---------------------------------------------------------------------------
__global__ __launch_bounds__(256) void pagtn_scatter_kernel(
    const int* __restrict__ src, const int* __restrict__ dst,
    const float* __restrict__ scores, const float* __restrict__ denom,
    const float* __restrict__ md, const float* __restrict__ medg,
    float* __restrict__ out, int EH)
{
  const int idx = blockIdx.x * blockDim.x + threadIdx.x;
  if (idx >= EH) return;
  const int e  = idx >> 2;
  const int hh = idx & 3;
  const int se = src[e], de = dst[e];
  const float alpha = scores[idx] / denom[de * 4 + hh];

  const float4* mp = (const float4*)(md + (size_t)(se * 4 + hh) * 64);
  const float4* gp = (const float4*)(medg + (size_t)e * 64);
  float* op = out + (size_t)(de * 4 + hh) * 64;
#pragma unroll 4
  for (int k = 0; k < 16; ++k) {
    const float4 a = mp[k], g = gp[k];
    atomicAdd(&op[k * 4 + 0], alpha * (a.x + g.x));
    atomicAdd(&op[k * 4 + 1], alpha * (a.y + g.y));
    atomicAdd(&op[k * 4 + 2], alpha * (a.z + g.z));
    atomicAdd(&op[k * 4 + 3], alpha * (a.w + g.w));
  }
}

// ---------------------------------------------------------------------------
extern "C" void kernel_launch(void* const* d_in, const int* in_sizes, int n_in,
                              void* d_out, int out_size, void* d_ws, size_t ws_size,
                              hipStream_t stream) {
  const float* node_feats = (const float*)d_in[0];   // (V,H,64)
  const float* edge_feats = (const float*)d_in[1];   // (E,64)
  const int*   src        = (const int*)d_in[2];
  const int*   dst        = (const int*)d_in[3];
  const float* Wsrc = (const float*)d_in[4];  const float* bsrc = (const float*)d_in[5];
  const float* Wdst = (const float*)d_in[6];  const float* bdst = (const float*)d_in[7];
  const float* Wedg = (const float*)d_in[8];  const float* bedg = (const float*)d_in[9];
  const float* Wdot = (const float*)d_in[10]; const float* bdot = (const float*)d_in[11];
  // d_in[12..13]: Wmsrc/bmsrc computed in reference but unused.
  const float* Wmdst = (const float*)d_in[14]; const float* bmdst = (const float*)d_in[15];
  const float* Wmedg = (const float*)d_in[16]; const float* bmedg = (const float*)d_in[17];
  const float* Wn    = (const float*)d_in[18]; const float* bn    = (const float*)d_in[19];

  const int F  = 64;
  const int H  = 4;
  const int E  = in_sizes[2];            // edge count
  const int Rn = in_sizes[0] / F;        // V*H rows
  const int EH = E * H;

  // Workspace layout (floats)
  float* ws    = (float*)d_ws;
  float* s_    = ws;                                 // Rn*64
  float* dprj_ = s_    + (size_t)Rn * F;             // Rn*64
  float* md_   = dprj_ + (size_t)Rn * F;             // Rn*64
  float* eatn_ = md_   + (size_t)Rn * F;             // E*64
  float* medg_ = eatn_ + (size_t)E * F;              // E*64
  float* sc_   = medg_ + (size_t)E * F;              // E*H
  float* mb_   = sc_   + (size_t)EH;                 // Rn
  float* dn_   = mb_   + (size_t)Rn;                 // Rn
  float* out   = (float*)d_out;                      // Rn*64

  // 1) Node projections: s, d, md to scratch; nout (+bias) straight into d_out.
  const int nbN = (Rn + 127) / 128;
  pagtn_proj64_kernel<<<nbN, 256, 0, stream>>>(node_feats, Rn,
      Wsrc, bsrc, s_, Wdst, bdst, dprj_, Wmdst, bmdst, md_, Wn, bn, out, 4);

  // 2) Edge projections: e_atn, medg.
  const int nbE = (E + 127) / 128;
  pagtn_proj64_kernel<<<nbE, 256, 0, stream>>>(edge_feats, E,
      Wedg, bedg, eatn_, Wmedg, bmedg, medg_,
      Wedg, bedg, eatn_, Wedg, bedg, eatn_, 2);

  // 3) Segment-max/-sum buffers.
  pagtn_init_kernel<<<(Rn + 255) / 256, 256, 0, stream>>>(mb_, dn_, Rn);

  // 4) Attention scores + segment max.
  pagtn_scores_kernel<<<(EH + 255) / 256, 256, 0, stream>>>(
      s_, dprj_, eatn_, src, dst, Wdot, bdot, sc_, mb_, EH);

  // 5) exp(score - max) + segment sum.
  pagtn_exsum_kernel<<<(EH + 255) / 256, 256, 0, stream>>>(dst, mb_, sc_, dn_, EH);

  // 6) alpha-weighted message scatter-add into d_out.
  pagtn_scatter_kernel<<<(EH + 255) / 256, 256, 0, stream>>>(
      src, dst, sc_, dn_, md_, medg_, out, EH);
}